// StateObsMLP_61821759259471
// MI455X (gfx1250) — compile-verified
//
#include <hip/hip_runtime.h>
#include <hip/hip_bf16.h>

typedef __attribute__((ext_vector_type(2))) float v2f;
typedef __attribute__((ext_vector_type(8))) float v8f;

#define BATCH 2048
#define NCAT 32
#define SDIM 64
#define EMB 256
#define NOUT 256
#define ODIM 512
#define HDIM 512           // 2*EMB
#define HSTR 516           // LDS stride for h tile (516 % 64 == 4 -> conflict-free A gather)
#define SSTR 68            // LDS stride for state tile

// ---------------------------------------------------------------------------
// Kernel A: bucket rows by category. counts[c], offsets[c] (exclusive scan),
// perm[] = row indices grouped by category. Final output values are
// order-invariant within a category, so atomic scatter order is harmless.
// ---------------------------------------------------------------------------
__global__ void bucket_by_cat(const int* __restrict__ cat_ids,
                              int* __restrict__ counts,
                              int* __restrict__ offsets,
                              int* __restrict__ perm) {
    __shared__ int s_cnt[NCAT];
    __shared__ int s_cur[NCAT];
    int t = threadIdx.x;
    if (t < NCAT) s_cnt[t] = 0;
    __syncthreads();
    for (int i = t; i < BATCH; i += blockDim.x)
        atomicAdd(&s_cnt[cat_ids[i]], 1);
    __syncthreads();
    if (t == 0) {
        int acc = 0;
        for (int c = 0; c < NCAT; ++c) {
            s_cur[c] = acc;
            offsets[c] = acc;
            counts[c] = s_cnt[c];
            acc += s_cnt[c];
        }
    }
    __syncthreads();
    for (int i = t; i < BATCH; i += blockDim.x) {
        int c = cat_ids[i];
        int pos = atomicAdd(&s_cur[c], 1);
        perm[pos] = i;
    }
}

// ---------------------------------------------------------------------------
// Kernel B: one block = (category c, tile of up to 16 rows of that category).
// 4 waves; each wave owns 4 consecutive 16-wide N-tiles per stage.
// fp32 WMMA 16x16x4 throughout (matches fp32 reference precision).
// ---------------------------------------------------------------------------
__global__ __launch_bounds__(128)
void moe_mlp_kernel(const float* __restrict__ state,
                    const float* __restrict__ obs,
                    const float* __restrict__ obs_W,
                    const float* __restrict__ obs_b,
                    const float* __restrict__ state_W,
                    const float* __restrict__ state_b,
                    const float* __restrict__ l2_W,
                    const float* __restrict__ l2_b,
                    const int* __restrict__ counts,
                    const int* __restrict__ offsets,
                    const int* __restrict__ perm,
                    float* __restrict__ out) {
    const int c    = blockIdx.y;
    const int tile = blockIdx.x;
    const int cnt  = counts[c];
    if (tile * 16 >= cnt) return;

    __shared__ int   sh_rows[16];
    __shared__ float sh_s[16 * SSTR];      // state tile, zero-padded rows
    __shared__ float sh_h[16 * HSTR];      // hidden = [relu(state_emb) | obs_emb]

    const int t    = threadIdx.x;
    const int wave = t >> 5;
    const int lane = t & 31;
    const int m    = lane & 15;            // A row / output N within tile
    const int half = lane >> 4;
    const int kb   = half * 2;             // K sub-offset for A/B fragments
    const int mb   = half * 8;             // M base for C/D fragments

    const int base = offsets[c];
    if (t < 16) {
        int idx = tile * 16 + t;
        sh_rows[t] = (idx < cnt) ? perm[base + idx] : -1;
    }
    __syncthreads();

    // Stage state tile into LDS (padded rows -> 0)
    for (int i = t; i < 16 * SDIM; i += 128) {
        int mm = i >> 6, kk = i & (SDIM - 1);
        int r = sh_rows[mm];
        sh_s[mm * SSTR + kk] = (r >= 0) ? state[(long)r * SDIM + kk] : 0.0f;
    }
    __syncthreads();

    const float* Ws = state_W + (long)c * SDIM * EMB;
    const float* bs = state_b + (long)c * EMB;
    const float* Wl = l2_W    + (long)c * HDIM * NOUT;
    const float* lb = l2_b    + (long)c * NOUT;

    // -------- Stage 1a: h[:, 0:256] = relu(state_tile @ Ws + bs) --------
    for (int j = 0; j < 4; ++j) {
        const int nt   = wave * 4 + j;       // N-tile 0..15
        const int ncol = nt * 16 + m;
        v8f acc = {};
        for (int k = 0; k < SDIM; k += 4) {
            v2f a, b;
            a.x = sh_s[m * SSTR + k + kb];
            a.y = sh_s[m * SSTR + k + kb + 1];
            b.x = Ws[(long)(k + kb)     * EMB + ncol];
            b.y = Ws[(long)(k + kb + 1) * EMB + ncol];
            acc = __builtin_amdgcn_wmma_f32_16x16x4_f32(
                false, a, false, b, (short)0, acc, false, false);
        }
        const float bias = bs[ncol];
        #pragma unroll
        for (int v = 0; v < 8; ++v)
            sh_h[(mb + v) * HSTR + ncol] = fmaxf(acc[v] + bias, 0.0f);
    }

    // -------- Stage 1b: h[:, 256:512] = obs_tile @ obs_W + obs_b --------
    {
        const int  r      = sh_rows[m];
        const float validf = (r >= 0) ? 1.0f : 0.0f;
        const long ro     = (long)((r >= 0) ? r : 0) * ODIM;
        for (int j = 0; j < 4; ++j) {
            const int nt   = wave * 4 + j;
            const int ncol = nt * 16 + m;
            v8f acc = {};
            for (int k = 0; k < ODIM; k += 4) {
                v2f a, b;
                a.x = obs[ro + k + kb]     * validf;
                a.y = obs[ro + k + kb + 1] * validf;
                b.x = obs_W[(long)(k + kb)     * EMB + ncol];
                b.y = obs_W[(long)(k + kb + 1) * EMB + ncol];
                acc = __builtin_amdgcn_wmma_f32_16x16x4_f32(
                    false, a, false, b, (short)0, acc, false, false);
            }
            const float bias = obs_b[ncol];
            #pragma unroll
            for (int v = 0; v < 8; ++v)
                sh_h[(mb + v) * HSTR + EMB + ncol] = acc[v] + bias;
        }
    }
    __syncthreads();

    // -------- Stage 2: out = h @ l2_W[c] + l2_b[c] --------
    for (int j = 0; j < 4; ++j) {
        const int nt   = wave * 4 + j;
        const int ncol = nt * 16 + m;
        v8f acc = {};
        for (int k = 0; k < HDIM; k += 4) {
            v2f a, b;
            a.x = sh_h[m * HSTR + k + kb];
            a.y = sh_h[m * HSTR + k + kb + 1];
            b.x = Wl[(long)(k + kb)     * NOUT + ncol];
            b.y = Wl[(long)(k + kb + 1) * NOUT + ncol];
            acc = __builtin_amdgcn_wmma_f32_16x16x4_f32(
                false, a, false, b, (short)0, acc, false, false);
        }
        const float bias = lb[ncol];
        #pragma unroll
        for (int v = 0; v < 8; ++v) {
            int r = sh_rows[mb + v];
            if (r >= 0)
                out[(long)r * NOUT + ncol] = acc[v] + bias;
        }
    }
}

extern "C" void kernel_launch(void* const* d_in, const int* in_sizes, int n_in,
                              void* d_out, int out_size, void* d_ws, size_t ws_size,
                              hipStream_t stream) {
    const float* state   = (const float*)d_in[0];
    const float* obs     = (const float*)d_in[1];
    const int*   cat_ids = (const int*)  d_in[2];
    const float* obs_W   = (const float*)d_in[3];
    const float* obs_b   = (const float*)d_in[4];
    const float* state_W = (const float*)d_in[5];
    const float* state_b = (const float*)d_in[6];
    const float* l2_W    = (const float*)d_in[7];
    const float* l2_b    = (const float*)d_in[8];
    float*       out     = (float*)d_out;

    int* counts  = (int*)d_ws;
    int* offsets = counts + NCAT;
    int* perm    = offsets + NCAT;

    bucket_by_cat<<<1, 256, 0, stream>>>(cat_ids, counts, offsets, perm);

    dim3 grid((BATCH + 15) / 16, NCAT);
    moe_mlp_kernel<<<grid, 128, 0, stream>>>(state, obs, obs_W, obs_b,
                                             state_W, state_b, l2_W, l2_b,
                                             counts, offsets, perm, out);
}